// GauntTensorProductAllParitiesS2Grid_28905129902790
// MI455X (gfx1250) — compile-verified
//
#include <hip/hip_runtime.h>
#include <hip/hip_bf16.h>
#include <math.h>

// ---------------------------------------------------------------------------
// GauntTensorProductAllParitiesS2Grid for MI455X (gfx1250), fp32 WMMA path.
// Four GEMM stages mapped to chained V_WMMA_F32_16X16X4_F32 (wave32).
// LDS tiles staged via GLOBAL_LOAD_ASYNC_TO_LDS (ASYNCcnt-tracked, no VGPR
// bounce); builtins probe-confirmed present on this toolchain.
// ---------------------------------------------------------------------------

typedef __attribute__((ext_vector_type(2))) float v2f;
typedef __attribute__((ext_vector_type(8))) float v8f;
typedef int vsi4 __attribute__((vector_size(16)));  // matches builtin's b128 pointee

static __device__ __forceinline__ v8f wmma_f32(v2f a, v2f b, v8f c) {
  // 8 args: (neg_a, A, neg_b, B, c_mod, C, reuse_a, reuse_b)
  return __builtin_amdgcn_wmma_f32_16x16x4_f32(false, a, false, b, (short)0, c,
                                               false, false);
}

#if defined(__has_builtin)
#if __has_builtin(__builtin_amdgcn_global_load_async_to_lds_b32) && \
    __has_builtin(__builtin_amdgcn_s_wait_asynccnt)
#define USE_ASYNC_LDS 1
#endif
#if __has_builtin(__builtin_amdgcn_global_load_async_to_lds_b128) && \
    __has_builtin(__builtin_amdgcn_s_wait_asynccnt)
#define USE_ASYNC_LDS_128 1
#endif
#endif

// Copy one f32 global -> LDS (per-lane), async when available.
static __device__ __forceinline__ void cp_f32_to_lds(const float* g, float* l) {
#ifdef USE_ASYNC_LDS
  __builtin_amdgcn_global_load_async_to_lds_b32(
      (__attribute__((address_space(1))) int*)g,
      (__attribute__((address_space(3))) int*)l, 0, 0);
#else
  *l = *g;
#endif
}

// Copy 16B global -> LDS (per-lane), async when available.
static __device__ __forceinline__ void cp_f32x4_to_lds(const float* g,
                                                       float* l) {
#ifdef USE_ASYNC_LDS_128
  __builtin_amdgcn_global_load_async_to_lds_b128(
      (__attribute__((address_space(1))) vsi4*)g,
      (__attribute__((address_space(3))) vsi4*)l, 0, 0);
#else
  *(float4*)l = *(const float4*)g;
#endif
}

// Wait for this wave's async LDS fills, then block barrier.
static __device__ __forceinline__ void lds_fill_barrier() {
#if defined(USE_ASYNC_LDS) || defined(USE_ASYNC_LDS_128)
  __builtin_amdgcn_s_wait_asynccnt(0);
#endif
  __syncthreads();
}

#define RSQRT128 0.08838834764831845f
#define TWO_PI 6.283185307179586f
#define SQRT2 1.4142135623730951f

// ---------------------------------------------------------------------------
// Kernel 0: build Yin (16 x 128) and Yow^T (128 x 64, cols 49..63 zeroed)
// on device. Gauss-Legendre(8) nodes/weights are compile-time constants.
// One thread per grid point g = beta*16 + alpha.
// ---------------------------------------------------------------------------
__global__ void k_shinit(float* __restrict__ yin, float* __restrict__ yowt) {
  const int g = threadIdx.x;            // 0..127
  const int bi = g >> 4;                // beta index 0..7
  const int ai = g & 15;                // alpha index 0..15
  const float nodes[8] = {-0.9602898564975363f, -0.7966664774136267f,
                          -0.5255324099163290f, -0.1834346424956498f,
                           0.1834346424956498f,  0.5255324099163290f,
                           0.7966664774136267f,  0.9602898564975363f};
  const float wts[8]   = { 0.1012285362903763f,  0.2223810344533745f,
                           0.3137066458778873f,  0.3626837833783620f,
                           0.3626837833783620f,  0.3137066458778873f,
                           0.2223810344533745f,  0.1012285362903763f};
  const float x = nodes[bi];
  const float alpha = TWO_PI * (float)ai / 16.0f;
  const float wq = wts[bi] * (TWO_PI / 16.0f);

  // Associated Legendre recurrence (matches reference _legendre), lmax = 6.
  float P[7][7];
  for (int i = 0; i < 7; ++i)
    for (int j = 0; j < 7; ++j) P[i][j] = 0.0f;
  const float s = sqrtf(fmaxf(0.0f, 1.0f - x * x));
  P[0][0] = 1.0f;
  for (int m = 1; m <= 6; ++m) P[m][m] = -(2.0f * m - 1.0f) * s * P[m - 1][m - 1];
  for (int m = 0; m < 6; ++m) P[m + 1][m] = (2.0f * m + 1.0f) * x * P[m][m];
  for (int m = 0; m <= 6; ++m)
    for (int l = m + 2; l <= 6; ++l)
      P[l][m] = ((2.0f * l - 1.0f) * x * P[l - 1][m] -
                 (float)(l - 1 + m) * P[l - 2][m]) / (float)(l - m);

  const float fact[13] = {1.f, 1.f, 2.f, 6.f, 24.f, 120.f, 720.f, 5040.f,
                          40320.f, 362880.f, 3628800.f, 39916800.f, 479001600.f};
  const float inv4pi = 0.07957747154594767f;
  for (int l = 0; l <= 6; ++l) {
    for (int m = -l; m <= l; ++m) {
      const int am = (m < 0) ? -m : m;
      const float norm =
          sqrtf((2.0f * l + 1.0f) * inv4pi * fact[l - am] / fact[l + am]);
      float az;
      if (m > 0)       az = SQRT2 * cosf((float)m * alpha);
      else if (m == 0) az = 1.0f;
      else             az = SQRT2 * sinf((float)am * alpha);
      const float Y = norm * P[l][am] * az;
      const int k = l * l + l + m;
      if (l <= 3) yin[k * 128 + g] = Y;      // Yin: [k][g]
      yowt[g * 64 + k] = Y * wq;             // Yow^T: [g][k'] (weighted)
    }
  }
  for (int k = 49; k < 64; ++k) yowt[g * 64 + k] = 0.0f;  // pad cols
}

// ---------------------------------------------------------------------------
// Kernel 1: project_in.  c[n][c][k] = (1/sqrt(MUL)) * sum_u X[n,u,pi(k),k]*W[l(k),u,c]
// grid = (N/16, 4 l-values); block = 256 (8 waves); wave w owns C-cols [16w,16w+16).
// X tile (16 x 128) staged in LDS per k; W fragments read straight from global
// (hot in L2: one 64KB slice per l reused by all blocks).
// ---------------------------------------------------------------------------
__global__ __launch_bounds__(256) void k_project(const float* __restrict__ X,
                                                 const float* __restrict__ W,
                                                 float* __restrict__ Cout,
                                                 int psel) {
  __shared__ float ldsA[16 * 128];  // 8 KB
  const int n0 = blockIdx.x * 16;
  const int l = blockIdx.y;         // 0..3
  const int tid = threadIdx.x;
  const int wave = tid >> 5;        // wave32: 8 waves
  const int lane = tid & 31;
  const int ln = lane & 15;
  const int hi = lane >> 4;         // 0: K/M lanes 0..15, 1: lanes 16..31
  const int c0 = wave * 16;
  const int pi = (l + psel) & 1;
  const float* __restrict__ Wl = W + l * (128 * 128);

  for (int k = l * l; k <= l * l + 2 * l; ++k) {
    // Stage A tile: A[m][u] = X[(((n0+m)*128 + u)*2 + pi)*16 + k]
    for (int idx = tid; idx < 16 * 128; idx += 256) {
      const int m = idx >> 7, u = idx & 127;
      cp_f32_to_lds(&X[(((n0 + m) * 128 + u) * 2 + pi) * 16 + k], &ldsA[idx]);
    }
    lds_fill_barrier();
    v8f acc = {};
#pragma unroll
    for (int t = 0; t < 32; ++t) {           // K = MUL = 128, 4 per WMMA
      const int u = 4 * t + hi * 2;
      const v2f a = *(const v2f*)&ldsA[ln * 128 + u];   // A[m=ln, u], A[m, u+1]
      v2f b;
      b.x = Wl[u * 128 + c0 + ln];            // B[u,   c]
      b.y = Wl[(u + 1) * 128 + c0 + ln];      // B[u+1, c]
      acc = wmma_f32(a, b, acc);
    }
#pragma unroll
    for (int j = 0; j < 8; ++j) {             // C/D: VGPR j -> row j (+8 for hi)
      const int m = j + hi * 8;
      Cout[((n0 + m) * 128 + c0 + ln) * 16 + k] = acc[j] * RSQRT128;
    }
    __syncthreads();
  }
}

// ---------------------------------------------------------------------------
// Kernel 2: fused grid stages.
//   g1 = c1 . Yin ; g2 = c2 . Yin ; P = g1 * g2 (in accumulator regs)
//   z  = P . Yow^T   (rows r = n*C + c; K' padded to 64)
// grid = (131072/32); block = 256 (8 waves). Wave w: row-tile w>>2, then
// g-tiles {w&3, (w&3)+4} for stage B and k'-tile (w&3) for stage C.
// ---------------------------------------------------------------------------
__global__ __launch_bounds__(256) void k_grid(const float* __restrict__ C1,
                                              const float* __restrict__ C2,
                                              const float* __restrict__ Yin,
                                              const float* __restrict__ YowT,
                                              float* __restrict__ Z) {
  __shared__ float ldsYin[16 * 128];   // 8 KB
  __shared__ float ldsYow[128 * 64];   // 32 KB
  __shared__ float ldsP[2][16 * 128];  // 16 KB
  const int tid = threadIdx.x;
  const int wave = tid >> 5;
  const int lane = tid & 31;
  const int ln = lane & 15;
  const int hi = lane >> 4;
  const int r0 = blockIdx.x * 32;
  const int rt = wave >> 2;
  const int rbase = r0 + rt * 16;

  // Contiguous constant tables: 16B-per-lane async fills.
  for (int idx = tid; idx < 512; idx += 256)
    cp_f32x4_to_lds(&Yin[idx * 4], &ldsYin[idx * 4]);
  for (int idx = tid; idx < 2048; idx += 256)
    cp_f32x4_to_lds(&YowT[idx * 4], &ldsYow[idx * 4]);
  lds_fill_barrier();

  // Stage B + elementwise product
  for (int gi = 0; gi < 2; ++gi) {
    const int g0 = ((wave & 3) + gi * 4) * 16;
    v8f acc1 = {}, acc2 = {};
#pragma unroll
    for (int t = 0; t < 4; ++t) {            // K = K_IN = 16
      const int kk = 4 * t + hi * 2;
      const v2f a1 = *(const v2f*)&C1[(rbase + ln) * 16 + kk];
      const v2f a2 = *(const v2f*)&C2[(rbase + ln) * 16 + kk];
      v2f b;
      b.x = ldsYin[kk * 128 + g0 + ln];
      b.y = ldsYin[(kk + 1) * 128 + g0 + ln];
      acc1 = wmma_f32(a1, b, acc1);
      acc2 = wmma_f32(a2, b, acc2);
    }
#pragma unroll
    for (int j = 0; j < 8; ++j) {            // product in registers -> LDS
      const int m = j + hi * 8;
      ldsP[rt][m * 128 + g0 + ln] = acc1[j] * acc2[j];
    }
  }
  __syncthreads();

  // Stage C: z[r, k'] = sum_g P[r,g] * Yow^T[g,k']
  const int kt = (wave & 3) * 16;
  v8f acc = {};
#pragma unroll
  for (int t = 0; t < 32; ++t) {             // K = G = 128
    const int g = 4 * t + hi * 2;
    const v2f a = *(const v2f*)&ldsP[rt][ln * 128 + g];
    v2f b;
    b.x = ldsYow[g * 64 + kt + ln];
    b.y = ldsYow[(g + 1) * 64 + kt + ln];
    acc = wmma_f32(a, b, acc);
  }
#pragma unroll
  for (int j = 0; j < 8; ++j) {
    const int m = j + hi * 8;
    Z[(rbase + m) * 64 + kt + ln] = acc[j];
  }
}

// ---------------------------------------------------------------------------
// Kernel 3: output projection.
//   o[n,d,k'] = (1/sqrt(C)) * sum_c z[n,c,k'] * Wout[l(k'),c,d]
// grid = (N/16, 49); block = 256 (8 waves); wave w owns d-cols [16w,16w+16).
// Final layout: out[n][p*6272 + d*49 + k'].
// ---------------------------------------------------------------------------
__global__ __launch_bounds__(256) void k_out(const float* __restrict__ Z,
                                             const float* __restrict__ Wo,
                                             float* __restrict__ Out, int p) {
  __shared__ float ldsA[16 * 128];  // 8 KB
  const int n0 = blockIdx.x * 16;
  const int k = blockIdx.y;         // 0..48
  int l = 0;
  while ((l + 1) * (l + 1) <= k) ++l;       // l(k') for LMAX_OUT = 6
  const float* __restrict__ Wl = Wo + l * (128 * 128);
  const int tid = threadIdx.x;
  const int wave = tid >> 5;
  const int lane = tid & 31;
  const int ln = lane & 15;
  const int hi = lane >> 4;
  const int d0 = wave * 16;

  for (int idx = tid; idx < 2048; idx += 256) {
    const int m = idx >> 7, c = idx & 127;
    cp_f32_to_lds(&Z[((n0 + m) * 128 + c) * 64 + k], &ldsA[idx]);
  }
  lds_fill_barrier();
  v8f acc = {};
#pragma unroll
  for (int t = 0; t < 32; ++t) {            // K = C = 128
    const int c = 4 * t + hi * 2;
    const v2f a = *(const v2f*)&ldsA[ln * 128 + c];
    v2f b;
    b.x = Wl[c * 128 + d0 + ln];
    b.y = Wl[(c + 1) * 128 + d0 + ln];
    acc = wmma_f32(a, b, acc);
  }
#pragma unroll
  for (int j = 0; j < 8; ++j) {
    const int m = j + hi * 8;
    Out[(n0 + m) * 18816 + p * 6272 + (d0 + ln) * 49 + k] = acc[j] * RSQRT128;
  }
}

// ---------------------------------------------------------------------------
// Launch: per-parity pipeline reusing workspace (~50 MB of floats).
// ---------------------------------------------------------------------------
extern "C" void kernel_launch(void* const* d_in, const int* in_sizes, int n_in,
                              void* d_out, int out_size, void* d_ws,
                              size_t ws_size, hipStream_t stream) {
  (void)in_sizes; (void)n_in; (void)out_size; (void)ws_size;
  const float* in1 = (const float*)d_in[0];   // (1024,128,2,16)
  const float* in2 = (const float*)d_in[1];
  const float* Win1 = (const float*)d_in[2];  // (3,4,128,128)
  const float* Win2 = (const float*)d_in[3];
  const float* Wout = (const float*)d_in[4];  // (3,7,128,128)
  float* out = (float*)d_out;                 // (1024, 3*128*49)
  float* ws = (float*)d_ws;

  float* yin  = ws;                 // 2048
  float* yowt = ws + 2048;          // 8192
  float* c1   = ws + 10240;         // 1024*128*16 = 2097152
  float* c2   = c1 + 2097152;       // 2097152
  float* z    = c2 + 2097152;       // 131072*64 = 8388608

  k_shinit<<<1, 128, 0, stream>>>(yin, yowt);

  // PARITY_CONFIGS = [(1,1),(1,-1),(-1,1)]; p=+1 -> psel 0, p=-1 -> psel 1
  const int psel1[3] = {0, 0, 1};
  const int psel2[3] = {0, 1, 0};
  for (int p = 0; p < 3; ++p) {
    k_project<<<dim3(64, 4), 256, 0, stream>>>(in1, Win1 + p * 65536, c1,
                                               psel1[p]);
    k_project<<<dim3(64, 4), 256, 0, stream>>>(in2, Win2 + p * 65536, c2,
                                               psel2[p]);
    k_grid<<<dim3(4096), 256, 0, stream>>>(c1, c2, yin, yowt, z);
    k_out<<<dim3(64, 49), 256, 0, stream>>>(z, Wout + p * 114688, out, p);
  }
}